// RobustGNN_77163382440209
// MI455X (gfx1250) — compile-verified
//
#include <hip/hip_runtime.h>
#include <hip/hip_bf16.h>

#define N_NODES 20000
#define N_EDGES 320000
#define HID     256
#define NGRAPH  64

typedef __attribute__((ext_vector_type(2))) float v2f;
typedef __attribute__((ext_vector_type(4))) float v4f;
typedef __attribute__((ext_vector_type(8))) float v8f;

// ----------------------------------------------------------------------------
// utility: zero int range
// ----------------------------------------------------------------------------
__global__ void gnn_zero_i32(int* __restrict__ p, int n) {
    int i = blockIdx.x * blockDim.x + threadIdx.x;
    if (i < n) p[i] = 0;
}

// ----------------------------------------------------------------------------
// in-degree histogram over dst (self-loop added later as +1)
// edge_index layout: ei[0..E-1] = src, ei[E..2E-1] = dst
// ----------------------------------------------------------------------------
__global__ void gnn_count_indeg(const int* __restrict__ ei, int* __restrict__ cnt, int E) {
    int e = blockIdx.x * blockDim.x + threadIdx.x;
    if (e < E) atomicAdd(&cnt[ei[E + e]], 1);
}

// dinv[i] = 1/sqrt(indeg + 1)   (self loop contributes 1, deg always > 0)
__global__ void gnn_make_dinv(const int* __restrict__ cnt, float* __restrict__ dinv, int n) {
    int i = blockIdx.x * blockDim.x + threadIdx.x;
    if (i < n) dinv[i] = 1.0f / sqrtf((float)(cnt[i] + 1));
}

// ----------------------------------------------------------------------------
// single-block exclusive scan of cnt[0..n-1] -> off[0..n]  (n ~ 20000, trivial)
// ----------------------------------------------------------------------------
__global__ void gnn_scan_excl(const int* __restrict__ cnt, int* __restrict__ off, int n) {
    __shared__ int sh[256];
    __shared__ int carry;
    int t = threadIdx.x;
    if (t == 0) carry = 0;
    __syncthreads();
    for (int base = 0; base < n; base += 256) {
        int v = (base + t < n) ? cnt[base + t] : 0;
        sh[t] = v;
        __syncthreads();
        for (int s = 1; s < 256; s <<= 1) {
            int add = (t >= s) ? sh[t - s] : 0;
            __syncthreads();
            sh[t] += add;
            __syncthreads();
        }
        if (base + t < n) off[base + t] = carry + sh[t] - v;
        __syncthreads();
        if (t == 255) carry += sh[255];
        __syncthreads();
    }
    if (t == 0) off[n] = carry;
}

// ----------------------------------------------------------------------------
// fill dst-sorted CSR: csr_s[p]=src, csr_w[p]=dinv[src]*dinv[dst]
// ----------------------------------------------------------------------------
__global__ void gnn_csr_fill(const int* __restrict__ ei, const int* __restrict__ off,
                             int* __restrict__ fill, const float* __restrict__ dinv,
                             int* __restrict__ csr_s, float* __restrict__ csr_w, int E) {
    int e = blockIdx.x * blockDim.x + threadIdx.x;
    if (e < E) {
        int s = ei[e];
        int d = ei[E + e];
        int p = off[d] + atomicAdd(&fill[d], 1);
        csr_s[p] = s;
        csr_w[p] = dinv[s] * dinv[d];
    }
}

// ----------------------------------------------------------------------------
// layer-1 GEMM: H = X(N,3) @ W1(3,256)   (K=3: scalar FMAs, store-bandwidth bound)
// ----------------------------------------------------------------------------
__global__ __launch_bounds__(HID)
void gnn_gemm_in3(const float* __restrict__ x, const float* __restrict__ W1,
                  float* __restrict__ H) {
    int i = blockIdx.x;
    int j = threadIdx.x;
    float x0 = x[i * 3 + 0], x1 = x[i * 3 + 1], x2 = x[i * 3 + 2];
    H[i * HID + j] = x0 * W1[j] + x1 * W1[HID + j] + x2 * W1[2 * HID + j];
}

// ----------------------------------------------------------------------------
// 256x256 GEMM with V_WMMA_F32_16X16X4_F32  (unchanged: known-good codegen)
// block = 256 threads (8 waves) -> 32 rows x 256 cols of H
// ----------------------------------------------------------------------------
#define GEMM_RT 2                 // row tiles (of 16) per block -> 32 rows
#define LDS_STRIDE 260            // pad: 16 A-lanes hit 16 distinct banks

__global__ __launch_bounds__(256)
void gnn_gemm256_wmma(const float* __restrict__ X, const float* __restrict__ W,
                      float* __restrict__ H) {
    __shared__ float sx[GEMM_RT * 16 * LDS_STRIDE];

    const int tid  = threadIdx.x;
    const int row0 = blockIdx.x * (GEMM_RT * 16);

    for (int idx = tid; idx < GEMM_RT * 16 * HID; idx += 256) {
        int r = idx >> 8;
        int c = idx & 255;
        sx[r * LDS_STRIDE + c] = X[(row0 + r) * HID + c];
    }
    __syncthreads();

    const int wave = tid >> 5;
    const int lane = tid & 31;
    const int col0 = wave * 32;
    const int lh   = lane & 15;            // N index / M index within fragment
    const int koff = (lane >> 4) << 1;     // lanes 0-15 -> K 0,1 ; 16-31 -> K 2,3

    v8f acc[GEMM_RT][2] = {};

    for (int k = 0; k < HID; k += 4) {
        const int kk = k + koff;
        v2f b0, b1;
        b0.x = W[kk * HID + col0 + lh];
        b0.y = W[(kk + 1) * HID + col0 + lh];
        b1.x = W[kk * HID + col0 + 16 + lh];
        b1.y = W[(kk + 1) * HID + col0 + 16 + lh];
#pragma unroll
        for (int r = 0; r < GEMM_RT; ++r) {
            v2f a = *(const v2f*)&sx[(r * 16 + lh) * LDS_STRIDE + kk];  // ds_load_b64
            acc[r][0] = __builtin_amdgcn_wmma_f32_16x16x4_f32(
                false, a, false, b0, (short)0, acc[r][0], false, false);
            acc[r][1] = __builtin_amdgcn_wmma_f32_16x16x4_f32(
                false, a, false, b1, (short)0, acc[r][1], false, false);
        }
    }

    const int mhalf = (lane >> 4) << 3;
#pragma unroll
    for (int r = 0; r < GEMM_RT; ++r) {
        int baseRow = row0 + r * 16 + mhalf;
#pragma unroll
        for (int q = 0; q < 8; ++q) {
            H[(baseRow + q) * HID + col0 + lh]      = acc[r][0][q];
            H[(baseRow + q) * HID + col0 + 16 + lh] = acc[r][1][q];
        }
    }
}

// ----------------------------------------------------------------------------
// aggregation (hot kernel): Xout[i] = relu( sum_e w_e*H[src_e] + dinv_i^2*H[i] + b )
// 256-thread block = 4 nodes; 64 lanes/node, float4 per lane -> b128 gathers.
// 2-deep software pipeline so two row-gathers are in flight per iteration.
// ----------------------------------------------------------------------------
__global__ __launch_bounds__(256)
void gnn_aggregate(const float* __restrict__ H, const int* __restrict__ off,
                   const int* __restrict__ csr_s, const float* __restrict__ csr_w,
                   const float* __restrict__ dinv, const float* __restrict__ bias,
                   float* __restrict__ Xout) {
    const int i = blockIdx.x * 4 + (threadIdx.x >> 6);   // node
    const int q = threadIdx.x & 63;                      // float4 slot
    const float di = dinv[i];

    const v4f hself = *(const v4f*)&H[(size_t)i * HID + q * 4];
    v4f acc = (di * di) * hself;                         // self loop

    int p = off[i];
    const int pe = off[i + 1];
    for (; p + 1 < pe; p += 2) {                         // 2 gathers in flight
        int   s0 = csr_s[p],     s1 = csr_s[p + 1];
        float w0 = csr_w[p],     w1 = csr_w[p + 1];
        v4f h0 = *(const v4f*)&H[(size_t)s0 * HID + q * 4];
        v4f h1 = *(const v4f*)&H[(size_t)s1 * HID + q * 4];
        acc += w0 * h0;
        acc += w1 * h1;
    }
    if (p < pe) {
        int   s0 = csr_s[p];
        float w0 = csr_w[p];
        acc += w0 * *(const v4f*)&H[(size_t)s0 * HID + q * 4];
    }

    const v4f b = *(const v4f*)&bias[q * 4];
    v4f r = acc + b;
    r.x = fmaxf(r.x, 0.0f);
    r.y = fmaxf(r.y, 0.0f);
    r.z = fmaxf(r.z, 0.0f);
    r.w = fmaxf(r.w, 0.0f);
    *(v4f*)&Xout[(size_t)i * HID + q * 4] = r;
}

// ----------------------------------------------------------------------------
// global_add_pool: batch is sorted; block g binary-searches its segment.
// 4 row-slices x 64 float4-lanes, LDS tree reduce. Deterministic, no atomics.
// ----------------------------------------------------------------------------
__device__ __forceinline__ int gnn_lowerb(const int* __restrict__ b, int n, int v) {
    int lo = 0, hi = n;
    while (lo < hi) { int m = (lo + hi) >> 1; if (b[m] < v) lo = m + 1; else hi = m; }
    return lo;
}

__global__ __launch_bounds__(256)
void gnn_pool(const float* __restrict__ X, const int* __restrict__ batch,
              float* __restrict__ pooled, int n) {
    __shared__ v4f red[4][64];
    const int g     = blockIdx.x;
    const int slice = threadIdx.x >> 6;     // 0..3
    const int q     = threadIdx.x & 63;     // float4 slot
    const int s = gnn_lowerb(batch, n, g);
    const int e = gnn_lowerb(batch, n, g + 1);
    v4f acc = {0.0f, 0.0f, 0.0f, 0.0f};
    for (int i = s + slice; i < e; i += 4)
        acc += *(const v4f*)&X[(size_t)i * HID + q * 4];
    red[slice][q] = acc;
    __syncthreads();
    if (slice == 0) {
        v4f r = red[0][q] + red[1][q] + red[2][q] + red[3][q];
        *(v4f*)&pooled[g * HID + q * 4] = r;
    }
}

// out[g,o] = pooled[g] . Wlin[:,o] + blin[o]
__global__ void gnn_head(const float* __restrict__ pooled, const float* __restrict__ Wlin,
                         const float* __restrict__ blin, float* __restrict__ out) {
    const int t = threadIdx.x;          // 0..127
    const int g = t >> 1, o = t & 1;
    float acc = blin[o];
    for (int j = 0; j < HID; ++j) acc += pooled[g * HID + j] * Wlin[j * 2 + o];
    out[g * 2 + o] = acc;
}

// ----------------------------------------------------------------------------
// launch
// ----------------------------------------------------------------------------
extern "C" void kernel_launch(void* const* d_in, const int* in_sizes, int n_in,
                              void* d_out, int out_size, void* d_ws, size_t ws_size,
                              hipStream_t stream) {
    const float* x     = (const float*)d_in[0];
    const int*   ei    = (const int*)d_in[1];
    const int*   batch = (const int*)d_in[2];
    const float* Wl[5] = { (const float*)d_in[3], (const float*)d_in[5],
                           (const float*)d_in[7], (const float*)d_in[9],
                           (const float*)d_in[11] };
    const float* bl[5] = { (const float*)d_in[4], (const float*)d_in[6],
                           (const float*)d_in[8], (const float*)d_in[10],
                           (const float*)d_in[12] };
    const float* Wlin = (const float*)d_in[13];
    const float* blin = (const float*)d_in[14];
    float* out = (float*)d_out;

    const int N = N_NODES, E = N_EDGES;

    // workspace layout (all 16-byte aligned rows; ~44 MB total)
    float* xcur   = (float*)d_ws;
    float* hbuf   = xcur + (size_t)N * HID;
    float* dinv   = hbuf + (size_t)N * HID;
    float* csr_w  = dinv + N;
    float* pooled = csr_w + E;
    int*   cnt    = (int*)(pooled + NGRAPH * HID);
    int*   fill   = cnt + N;
    int*   off    = fill + N;            // N+1 entries
    int*   csr_s  = off + N + 1;

    const int T = 256;

    // --- build normalization + dst-sorted CSR (once per launch) ---
    gnn_zero_i32<<<(2 * N + T - 1) / T, T, 0, stream>>>(cnt, 2 * N);   // cnt + fill
    gnn_count_indeg<<<(E + T - 1) / T, T, 0, stream>>>(ei, cnt, E);
    gnn_make_dinv<<<(N + T - 1) / T, T, 0, stream>>>(cnt, dinv, N);
    gnn_scan_excl<<<1, 256, 0, stream>>>(cnt, off, N);
    gnn_csr_fill<<<(E + T - 1) / T, T, 0, stream>>>(ei, off, fill, dinv, csr_s, csr_w, E);

    // --- layer 1 (K=3 GEMM) ---
    gnn_gemm_in3<<<N, HID, 0, stream>>>(x, Wl[0], hbuf);
    gnn_aggregate<<<N / 4, 256, 0, stream>>>(hbuf, off, csr_s, csr_w, dinv, bl[0], xcur);

    // --- layers 2..5 (WMMA f32 GEMM + vectorized gather aggregation) ---
    const int gemmBlocks = N / (GEMM_RT * 16);     // 20000 / 32 = 625
    for (int l = 1; l < 5; ++l) {
        gnn_gemm256_wmma<<<gemmBlocks, 256, 0, stream>>>(xcur, Wl[l], hbuf);
        gnn_aggregate<<<N / 4, 256, 0, stream>>>(hbuf, off, csr_s, csr_w, dinv, bl[l], xcur);
    }

    // --- pool + head ---
    gnn_pool<<<NGRAPH, 256, 0, stream>>>(xcur, batch, pooled, N);
    gnn_head<<<1, 128, 0, stream>>>(pooled, Wlin, blin, out);

    (void)in_sizes; (void)n_in; (void)out_size; (void)ws_size;
}